// StochasticTwoLayerGCN_4793183502743
// MI455X (gfx1250) — compile-verified
//
#include <hip/hip_runtime.h>

// ---------------------------------------------------------------------------
// Two-layer GCN for MI455X (gfx1250, wave32).
//   layer(h) = relu( (segsum_{e:dst}  h[src_e]*norm_src[src_e]) * norm_dst @ W + b )
// SpMM is L2-atomic bound (entire working set < 192MB L2); the dense
// 50000x128 @ 128x128 GEMM uses V_WMMA_F32_16X16X4_F32 (exact fp32 math).
// ---------------------------------------------------------------------------

#define N_NODES 50000
#define N_EDGES 600000
#define D       128

typedef __attribute__((ext_vector_type(2))) float v2f;
typedef __attribute__((ext_vector_type(8))) float v8f;

// ---------------------------------------------------------------- utilities
__global__ void zero_f32(float* __restrict__ p, int n) {
  int i = blockIdx.x * blockDim.x + threadIdx.x;
  if (i < n) p[i] = 0.0f;
}

// one thread per edge: accumulate float degrees
__global__ void degree_kernel(const int* __restrict__ src,
                              const int* __restrict__ dst,
                              float* __restrict__ deg_out,
                              float* __restrict__ deg_in) {
  int e = blockIdx.x * blockDim.x + threadIdx.x;
  if (e >= N_EDGES) return;
  atomicAdd(&deg_out[src[e]], 1.0f);
  atomicAdd(&deg_in [dst[e]], 1.0f);
}

__global__ void norm_kernel(const float* __restrict__ deg_out,
                            const float* __restrict__ deg_in,
                            float* __restrict__ norm_src,
                            float* __restrict__ norm_dst) {
  int i = blockIdx.x * blockDim.x + threadIdx.x;
  if (i >= N_NODES) return;
  norm_src[i] = rsqrtf(fmaxf(deg_out[i], 1.0f));
  norm_dst[i] = rsqrtf(fmaxf(deg_in [i], 1.0f));
}

// ------------------------------------------------------------------- SpMM
// One wave (32 lanes) per edge; lane handles 4 consecutive floats (D=128).
// float4 gather of the source row (coalesced 512B per edge), scaled by
// norm_src, scattered with global_atomic_add_f32 into agg[dst].
__global__ __launch_bounds__(256)
void scatter_kernel(const float* __restrict__ h,
                    const float* __restrict__ norm_src,
                    const int*   __restrict__ src,
                    const int*   __restrict__ dst,
                    float*       __restrict__ agg) {
  int tid  = blockIdx.x * blockDim.x + threadIdx.x;
  int e    = tid >> 5;
  int lane = tid & 31;
  if (e >= N_EDGES) return;
  int s = src[e];
  int d = dst[e];
  float ns = norm_src[s];
  float4 v = ((const float4*)(h + (size_t)s * D))[lane];
  float* ap = agg + (size_t)d * D + lane * 4;
  atomicAdd(ap + 0, v.x * ns);
  atomicAdd(ap + 1, v.y * ns);
  atomicAdd(ap + 2, v.z * ns);
  atomicAdd(ap + 3, v.w * ns);
}

// ----------------------------------------------------------- WMMA GEMM
// out = relu( (agg * norm_dst[:,None]) @ W + b ),  M=50000, N=128, K=128.
// One wave per 16x16 output tile; K consumed in 4-wide steps with
// v_wmma_f32_16x16x4_f32 (fp32 in, fp32 accumulate).
//
// Operand layout (ISA 7.12.2, wave32):
//   A 16x4 f32 : lane L -> row M=L&15 ; VGPR{0,1} hold K = koff, koff+1
//                with koff = (L>>4)*2  (lanes 16-31 carry K+2/K+3)
//   B 4x16 f32 : lane L -> col N=L&15 ; same K split across the lane halves
//   C/D 16x16  : VGPR v, lane L -> M = v + (L>>4)*8, N = L&15
__global__ __launch_bounds__(256)
void gemm_bias_relu_wmma(const float* __restrict__ agg,
                         const float* __restrict__ norm_dst,
                         const float* __restrict__ W,
                         const float* __restrict__ bias,
                         float*       __restrict__ out) {
  int wave = (blockIdx.x * blockDim.x + threadIdx.x) >> 5;
  int lane = threadIdx.x & 31;
  int mt = wave >> 3;          // 0 .. 3124   (50000/16 M-tiles)
  int nt = wave & 7;           // 0 .. 7      (128/16   N-tiles)
  int m0 = mt * 16;
  int n0 = nt * 16;

  int row  = lane & 15;        // M within tile for A
  int col  = lane & 15;        // N within tile for B / C / D
  int koff = (lane >> 4) * 2;  // K sub-offset carried by lane half

  float nd = norm_dst[m0 + row];
  const float* arow = agg + (size_t)(m0 + row) * D;

  v8f c = {};
  #pragma unroll 8
  for (int k0 = 0; k0 < D; k0 += 4) {
    v2f a, b;
    a.x = arow[k0 + koff]     * nd;
    a.y = arow[k0 + koff + 1] * nd;
    b.x = W[(k0 + koff)     * D + n0 + col];
    b.y = W[(k0 + koff + 1) * D + n0 + col];
    // 8 args: (neg_a, A, neg_b, B, c_mod, C, reuse_a, reuse_b)
    c = __builtin_amdgcn_wmma_f32_16x16x4_f32(
            false, a, false, b, (short)0, c, false, false);
  }

  float bb = bias[n0 + col];
  int mbase = m0 + (lane >> 4) * 8;
  #pragma unroll
  for (int v = 0; v < 8; ++v) {
    float r = c[v] + bb;
    out[(size_t)(mbase + v) * D + n0 + col] = fmaxf(r, 0.0f);
  }
}

// --------------------------------------------------------------- launcher
extern "C" void kernel_launch(void* const* d_in, const int* in_sizes, int n_in,
                              void* d_out, int out_size, void* d_ws, size_t ws_size,
                              hipStream_t stream) {
  const float* x   = (const float*)d_in[0];
  const int*   src = (const int*)  d_in[1];
  const int*   dst = (const int*)  d_in[2];
  const float* W1  = (const float*)d_in[3];
  const float* b1  = (const float*)d_in[4];
  const float* W2  = (const float*)d_in[5];
  const float* b2  = (const float*)d_in[6];
  float* out = (float*)d_out;

  // workspace carve-up (all float, 16B-aligned sections)
  char* ws = (char*)d_ws;
  float* deg_out  = (float*)ws;  ws += (size_t)N_NODES * sizeof(float);
  float* deg_in   = (float*)ws;  ws += (size_t)N_NODES * sizeof(float);
  float* norm_src = (float*)ws;  ws += (size_t)N_NODES * sizeof(float);
  float* norm_dst = (float*)ws;  ws += (size_t)N_NODES * sizeof(float);
  float* agg      = (float*)ws;  ws += (size_t)N_NODES * D * sizeof(float);
  float* h1       = (float*)ws;  ws += (size_t)N_NODES * D * sizeof(float);

  const int T = 256;
  // degrees + norms (deg_out/deg_in are contiguous -> one zero pass)
  zero_f32<<<(2 * N_NODES + T - 1) / T, T, 0, stream>>>(deg_out, 2 * N_NODES);
  degree_kernel<<<(N_EDGES + T - 1) / T, T, 0, stream>>>(src, dst, deg_out, deg_in);
  norm_kernel<<<(N_NODES + T - 1) / T, T, 0, stream>>>(deg_out, deg_in, norm_src, norm_dst);

  const int scatter_blocks = (N_EDGES * 32 + T - 1) / T;          // 75000
  const int gemm_blocks    = ((N_NODES / 16) * 8 * 32) / T;       // 3125
  const int agg_elems      = N_NODES * D;

  // ---- layer 1: x -> h1
  zero_f32<<<(agg_elems + T - 1) / T, T, 0, stream>>>(agg, agg_elems);
  scatter_kernel<<<scatter_blocks, T, 0, stream>>>(x, norm_src, src, dst, agg);
  gemm_bias_relu_wmma<<<gemm_blocks, T, 0, stream>>>(agg, norm_dst, W1, b1, h1);

  // ---- layer 2: h1 -> out
  zero_f32<<<(agg_elems + T - 1) / T, T, 0, stream>>>(agg, agg_elems);
  scatter_kernel<<<scatter_blocks, T, 0, stream>>>(h1, norm_src, src, dst, agg);
  gemm_bias_relu_wmma<<<gemm_blocks, T, 0, stream>>>(agg, norm_dst, W2, b2, out);
}